// EdgeGatedGraphConv_74371653697786
// MI455X (gfx1250) — compile-verified
//
#include <hip/hip_runtime.h>
#include <math.h>

typedef float v2f __attribute__((ext_vector_type(2)));
typedef float v8f __attribute__((ext_vector_type(8)));

#define D64 64
// LDS layout for W: pair-interleaved. Pair-row p (K rows 2p,2p+1), pitch 160
// floats (160 % 64 == 32 -> halves of the wave hit disjoint bank sets).
// Wl[p*160 + c*2 + 0] = W[2p][c], Wl[p*160 + c*2 + 1] = W[2p+1][c]
#define WPITCH 160
#define WLDS (32 * WPITCH)  // 5120 floats = 20KB

__device__ __forceinline__ float sigm(float x) { return 1.0f / (1.0f + __expf(-x)); }

// ---------------------------------------------------------------------------
// Kernel 0: zero the segment-sum accumulators
// ---------------------------------------------------------------------------
__global__ void zero_kernel(float* p, int count) {
    int i = blockIdx.x * blockDim.x + threadIdx.x;
    if (i < count) p[i] = 0.0f;
}

// ---------------------------------------------------------------------------
// Kernel 1: W = w1[64,16] @ w2[16,64]  (one launch per layer; 4096 threads)
// ---------------------------------------------------------------------------
__global__ void build_w_kernel(const float* __restrict__ w1,
                               const float* __restrict__ w2,
                               float* __restrict__ Wout) {
    int idx = blockIdx.x * blockDim.x + threadIdx.x; // 0..4095
    int i = idx >> 6, j = idx & 63;
    float s = 0.0f;
#pragma unroll
    for (int r = 0; r < 16; ++r) s += w1[i * 16 + r] * w2[r * 64 + j];
    Wout[idx] = s;
}

// stage W[64,64] (row-major) into pair-interleaved LDS layout
__device__ __forceinline__ void stage_w(float* Wl, const float* __restrict__ Wm,
                                        int tid) {
    for (int i = tid; i < 4096; i += 256) {
        const int row = i >> 6, col = i & 63;
        Wl[(row >> 1) * WPITCH + col * 2 + (row & 1)] = Wm[i];
    }
}

// ---------------------------------------------------------------------------
// Kernel 2: node GEMMs via WMMA f32 16x16x4
//   out[rows,64] = node_feats[rows,64] @ W[64,64] + b   (4 layers: sg,dg,su,du)
// One wave = one 16-row tile. Block = 8 waves. W staged in LDS per layer.
// ---------------------------------------------------------------------------
__global__ __launch_bounds__(256) void node_gemm_kernel(
    const float* __restrict__ nf, const float* __restrict__ Wbase,
    const float* __restrict__ sgb, const float* __restrict__ dgb,
    const float* __restrict__ sub, const float* __restrict__ dub,
    float* __restrict__ esrc, float* __restrict__ edst,
    float* __restrict__ sux, float* __restrict__ bh, int N) {
    __shared__ float Wl[WLDS];
    const int tid = threadIdx.x;
    const int wave = tid >> 5, lane = tid & 31;
    const int half = lane >> 4, l15 = lane & 15;
    const int tile = blockIdx.x * 8 + wave;
    const int row0 = tile * 16;

    const int    msel[4] = {0, 1, 3, 4};  // sg, dg, su, du in W buffer
    const float* bsel[4] = {sgb, dgb, sub, dub};
    float*       osel[4] = {esrc, edst, sux, bh};

    for (int mat = 0; mat < 4; ++mat) {
        stage_w(Wl, Wbase + msel[mat] * 4096, tid);
        __syncthreads();

        if (row0 < N) {
            v8f acc[4];
#pragma unroll
            for (int nt = 0; nt < 4; ++nt)
#pragma unroll
                for (int g = 0; g < 8; ++g) acc[nt][g] = 0.0f;

            const float* arow = nf + (size_t)(row0 + l15) * D64;
#pragma unroll
            for (int k = 0; k < 16; ++k) {
                const int kk = 4 * k + 2 * half;   // even
                const int p = 2 * k + half;        // pair-row in LDS
                const v2f a = *(const v2f*)(arow + kk);
#pragma unroll
                for (int nt = 0; nt < 4; ++nt) {
                    const int c = nt * 16 + l15;
                    const v2f b = *(const v2f*)(&Wl[p * WPITCH + c * 2]);
                    acc[nt] = __builtin_amdgcn_wmma_f32_16x16x4_f32(
                        false, a, false, b, (short)0, acc[nt], false, false);
                }
            }
            float* out = osel[mat];
            const float* bias = bsel[mat];
#pragma unroll
            for (int g = 0; g < 8; ++g) {
                const int r = g + 8 * half;
                const size_t ro = (size_t)(row0 + r) * D64;
#pragma unroll
                for (int nt = 0; nt < 4; ++nt) {
                    const int c = nt * 16 + l15;
                    out[ro + c] = acc[nt][g] + bias[c];
                }
            }
        }
        __syncthreads();
    }
}

// ---------------------------------------------------------------------------
// Kernel 3: fused edge pipeline
//   e_edge = edge_feats @ W_eg + b (WMMA)
//   m = e_src[src] + e_dst[dst] + e_edge ; sigma = sigmoid(m)
//   atomic segment sums; y = edge_feats + silu(LN(m))
// ---------------------------------------------------------------------------
__global__ __launch_bounds__(256) void edge_kernel(
    const float* __restrict__ ef, const int* __restrict__ src,
    const int* __restrict__ dst, const float* __restrict__ Weg,
    const float* __restrict__ egb, const float* __restrict__ esrc,
    const float* __restrict__ edst, const float* __restrict__ bh,
    const float* __restrict__ lng, const float* __restrict__ lnb,
    float* __restrict__ ssh, float* __restrict__ ssum,
    float* __restrict__ yout, int E) {
    __shared__ float Wl[WLDS];
    const int tid = threadIdx.x;
    stage_w(Wl, Weg, tid);
    __syncthreads();

    const int wave = tid >> 5, lane = tid & 31;
    const int half = lane >> 4, l15 = lane & 15;
    const int tile = blockIdx.x * 8 + wave;
    const int row0 = tile * 16;
    if (row0 >= E) return;  // wave-uniform; no barriers below

    v8f acc[4];
#pragma unroll
    for (int nt = 0; nt < 4; ++nt)
#pragma unroll
        for (int g = 0; g < 8; ++g) acc[nt][g] = 0.0f;

    const float* arow = ef + (size_t)(row0 + l15) * D64;
#pragma unroll
    for (int k = 0; k < 16; ++k) {
        const int kk = 4 * k + 2 * half;
        const int p = 2 * k + half;
        const v2f a = *(const v2f*)(arow + kk);
#pragma unroll
        for (int nt = 0; nt < 4; ++nt) {
            const int c = nt * 16 + l15;
            const v2f b = *(const v2f*)(&Wl[p * WPITCH + c * 2]);
            acc[nt] = __builtin_amdgcn_wmma_f32_16x16x4_f32(
                false, a, false, b, (short)0, acc[nt], false, false);
        }
    }

    // per-16-edge tile: src/dst indices in lanes 0..15 (replicated in 16..31)
    const int sv = src[row0 + l15];
    const int dv = dst[row0 + l15];

    // m, sigma, atomic segment sums; keep m in acc
#pragma unroll
    for (int g = 0; g < 8; ++g) {
        const int r = g + 8 * half;
        const int s = __shfl(sv, r, 32);
        const int d = __shfl(dv, r, 32);
#pragma unroll
        for (int nt = 0; nt < 4; ++nt) {
            const int c = nt * 16 + l15;
            float m = acc[nt][g] + egb[c] + esrc[(size_t)s * D64 + c] +
                      edst[(size_t)d * D64 + c];
            const float sg = sigm(m);
            atomicAdd(&ssh[(size_t)d * D64 + c], bh[(size_t)s * D64 + c] * sg);
            atomicAdd(&ssum[(size_t)d * D64 + c], sg);
            acc[nt][g] = m;
        }
    }

    // LayerNorm(m) over D=64 per row (16-lane butterfly), SiLU, residual -> y
#pragma unroll
    for (int g = 0; g < 8; ++g) {
        const int r = g + 8 * half;
        float sum = acc[0][g] + acc[1][g] + acc[2][g] + acc[3][g];
        float sq = acc[0][g] * acc[0][g] + acc[1][g] * acc[1][g] +
                   acc[2][g] * acc[2][g] + acc[3][g] * acc[3][g];
#pragma unroll
        for (int off = 1; off < 16; off <<= 1) {
            sum += __shfl_xor(sum, off, 32);
            sq += __shfl_xor(sq, off, 32);
        }
        const float mu = sum * (1.0f / 64.0f);
        float var = sq * (1.0f / 64.0f) - mu * mu;
        var = var < 0.0f ? 0.0f : var;
        const float rs = rsqrtf(var + 1e-5f);
        const size_t ro = (size_t)(row0 + r) * D64;
#pragma unroll
        for (int nt = 0; nt < 4; ++nt) {
            const int c = nt * 16 + l15;
            float v = (acc[nt][g] - mu) * rs;
            v = v * lng[c] + lnb[c];
            v = v * sigm(v);  // silu
            yout[ro + c] = ef[ro + c] + v;
        }
    }
}

// ---------------------------------------------------------------------------
// Kernel 4: node finalize. One wave per node row (2 cols/lane).
// ---------------------------------------------------------------------------
__global__ __launch_bounds__(256) void node_final_kernel(
    const float* __restrict__ nf, const float* __restrict__ sux,
    const float* __restrict__ ssh, const float* __restrict__ ssum,
    const float* __restrict__ lng, const float* __restrict__ lnb,
    float* __restrict__ xout, int N) {
    const int wave = threadIdx.x >> 5, lane = threadIdx.x & 31;
    const int row = blockIdx.x * 8 + wave;
    if (row >= N) return;
    const size_t off = (size_t)row * D64;
    const int c0 = lane, c1 = lane + 32;
    float x0 = sux[off + c0] + ssh[off + c0] / (ssum[off + c0] + 1e-6f);
    float x1 = sux[off + c1] + ssh[off + c1] / (ssum[off + c1] + 1e-6f);
    float sum = x0 + x1;
    float sq = x0 * x0 + x1 * x1;
#pragma unroll
    for (int o = 1; o < 32; o <<= 1) {
        sum += __shfl_xor(sum, o, 32);
        sq += __shfl_xor(sq, o, 32);
    }
    const float mu = sum * (1.0f / 64.0f);
    float var = sq * (1.0f / 64.0f) - mu * mu;
    var = var < 0.0f ? 0.0f : var;
    const float rs = rsqrtf(var + 1e-5f);
    float v0 = ((x0 - mu) * rs) * lng[c0] + lnb[c0];
    float v1 = ((x1 - mu) * rs) * lng[c1] + lnb[c1];
    v0 = v0 * sigm(v0);
    v1 = v1 * sigm(v1);
    xout[off + c0] = nf[off + c0] + v0;
    xout[off + c1] = nf[off + c1] + v1;
}

// ---------------------------------------------------------------------------
extern "C" void kernel_launch(void* const* d_in, const int* in_sizes, int n_in,
                              void* d_out, int out_size, void* d_ws, size_t ws_size,
                              hipStream_t stream) {
    const float* nf = (const float*)d_in[0];
    const float* ef = (const float*)d_in[1];
    const int* src = (const int*)d_in[2];
    const int* dst = (const int*)d_in[3];
    // per-layer weights: [4 + 3*l] = w1, [5 + 3*l] = w2, [6 + 3*l] = b
    const float* sg_b = (const float*)d_in[6];
    const float* dg_b = (const float*)d_in[9];
    const float* eg_b = (const float*)d_in[12];
    const float* su_b = (const float*)d_in[15];
    const float* du_b = (const float*)d_in[18];
    const float* ln_n_g = (const float*)d_in[19];
    const float* ln_n_b = (const float*)d_in[20];
    const float* ln_e_g = (const float*)d_in[21];
    const float* ln_e_b = (const float*)d_in[22];

    const int N = in_sizes[0] / D64;
    const int E = in_sizes[1] / D64;

    // workspace layout (floats)
    float* W = (float*)d_ws;                 // 5 * 4096
    float* esrc = W + 5 * 4096;              // N*64
    float* edst = esrc + (size_t)N * D64;    // N*64
    float* bh = edst + (size_t)N * D64;      // N*64  (du)
    float* sux = bh + (size_t)N * D64;       // N*64  (su)
    float* ssh = sux + (size_t)N * D64;      // N*64  accum
    float* ssum = ssh + (size_t)N * D64;     // N*64  accum

    float* xout = (float*)d_out;             // [N,64]
    float* yout = xout + (size_t)N * D64;    // [E,64]

    // 0) zero accumulators
    {
        int cnt = 2 * N * D64;
        zero_kernel<<<(cnt + 255) / 256, 256, 0, stream>>>(ssh, cnt);
    }
    // 1) combined weights W_l = w1_l @ w2_l   (order: sg, dg, eg, su, du)
    for (int l = 0; l < 5; ++l) {
        build_w_kernel<<<16, 256, 0, stream>>>(
            (const float*)d_in[4 + 3 * l], (const float*)d_in[5 + 3 * l],
            W + l * 4096);
    }
    // 2) node GEMMs (sg->esrc, dg->edst, su->sux, du->bh)
    {
        int ntiles = N / 16;
        node_gemm_kernel<<<(ntiles + 7) / 8, 256, 0, stream>>>(
            nf, W, sg_b, dg_b, su_b, du_b, esrc, edst, sux, bh, N);
    }
    // 3) fused edge pipeline
    {
        int etiles = E / 16;
        edge_kernel<<<(etiles + 7) / 8, 256, 0, stream>>>(
            ef, src, dst, W + 2 * 4096, eg_b, esrc, edst, bh, ln_e_g, ln_e_b,
            ssh, ssum, yout, E);
    }
    // 4) node finalize
    node_final_kernel<<<(N + 7) / 8, 256, 0, stream>>>(
        nf, sux, ssh, ssum, ln_n_g, ln_n_b, xout, N);
}